// IF_with_neg_18622978195597
// MI455X (gfx1250) — compile-verified
//
#include <hip/hip_runtime.h>

// IF neuron with negative spikes, T=8 timesteps.
// x: [T*B, C, H, W] fp32 viewed as T planes of N = B*C*H*W elements.
// Per element: sequential T-step recurrence on (mem, cum); out[t] = spike.
//
// Memory-bound (0.25 FLOP/byte): 205 MB traffic @ 23.3 TB/s => ~8.8 us floor.
// Strategy: 1 lane owns 4 consecutive elements (b128), preloads all 8
// timestep tiles non-temporally (8 outstanding global_load_b128), runs the
// branchless recurrence in registers, streams 8 non-temporal b128 stores.

#define T_STEPS 8

// Native clang vector type: required by __builtin_nontemporal_* (HIP float4
// is a struct and is rejected). Same 16B layout -> b128 vmem ops.
typedef float v4f __attribute__((ext_vector_type(4)));

__global__ __launch_bounds__(256) void if_neg_spike_kernel(
    const float* __restrict__ x,
    const float* __restrict__ thresh_p,
    float* __restrict__ out,
    int n4)  // v4f elements per timestep plane
{
  const int i = blockIdx.x * blockDim.x + threadIdx.x;
  if (i >= n4) return;

  const float th = thresh_p[0];  // uniform -> s_load_b32

  const v4f* __restrict__ xv = reinterpret_cast<const v4f*>(x);
  v4f* __restrict__ ov = reinterpret_cast<v4f*>(out);

  // Preload all T timestep tiles: 8 outstanding b128 NT loads per lane.
  v4f xs[T_STEPS];
#pragma unroll
  for (int t = 0; t < T_STEPS; ++t) {
    xs[t] = __builtin_nontemporal_load(&xv[(size_t)t * (size_t)n4 + (size_t)i]);
  }

  float mem[4], cum[4];
#pragma unroll
  for (int k = 0; k < 4; ++k) {
    mem[k] = 0.5f * th;
    cum[k] = 0.0f;
  }

#pragma unroll
  for (int t = 0; t < T_STEPS; ++t) {
    v4f o;
#pragma unroll
    for (int k = 0; k < 4; ++k) {
      float m = mem[k] + xs[t][k];
      // spike     = zif(mem - th) * th             -> (m >= th) ? th : 0
      // spike_neg = zif(-mem) * (cum - th >= 0)*th -> (m <= 0 && cum >= th) ? th : 0
      // (mutually exclusive since th > 0)
      float s  = (m >= th) ? th : 0.0f;
      float sn = ((m <= 0.0f) && (cum[k] >= th)) ? th : 0.0f;
      float spike = s - sn;
      cum[k] = cum[k] + spike;
      mem[k] = m - spike;
      o[k] = spike;
    }
    __builtin_nontemporal_store(o, &ov[(size_t)t * (size_t)n4 + (size_t)i]);
  }
}

extern "C" void kernel_launch(void* const* d_in, const int* in_sizes, int n_in,
                              void* d_out, int out_size, void* d_ws, size_t ws_size,
                              hipStream_t stream) {
  (void)n_in; (void)out_size; (void)d_ws; (void)ws_size;

  const float* x = (const float*)d_in[0];
  const float* thresh = (const float*)d_in[1];
  float* out = (float*)d_out;

  const int n_total = in_sizes[0];            // T * B * C * H * W
  const int n_per_t = n_total / T_STEPS;      // elements per timestep plane
  const int n4 = n_per_t / 4;                 // v4f per plane (196*512*32 % 4 == 0)

  const int block = 256;                      // 8 wave32 waves per workgroup
  const int grid = (n4 + block - 1) / block;  // 3136 blocks

  if_neg_spike_kernel<<<grid, block, 0, stream>>>(x, thresh, out, n4);
}